// GraphConvolution_45870250721671
// MI455X (gfx1250) — compile-verified
//
#include <hip/hip_runtime.h>

typedef __attribute__((ext_vector_type(2))) float v2f;
typedef __attribute__((ext_vector_type(4))) float v4f;
typedef __attribute__((ext_vector_type(8))) float v8f;

#define D 256
#define TILE_K 32
#define AS_STRIDE 36   // 16 rows x 32 K (+pad): 36*m mod 64 distinct, even -> b64-aligned, conflict-free
#define BS_STRIDE 264  // 32 rows x 256 N (+pad): 2*264 mod 64 = 16 -> half-waves hit disjoint banks

// ---------------------------------------------------------------------------
// GEMM: support = x @ weight  using V_WMMA_F32_16X16X4_F32 (fp32-exact path).
// block = 256 threads = 8 waves; block tile = 16(M) x 256(N); K staged by 32.
// Double-buffered LDS + register prefetch: global loads for chunk k+1 are
// issued before the WMMA sequence of chunk k. X loads are non-temporal so the
// single-use 102 MB stream does not evict support/out from the 192 MB L2.
// ---------------------------------------------------------------------------
__global__ __launch_bounds__(256) void gcn_gemm_wmma(const float* __restrict__ X,
                                                     const float* __restrict__ W,
                                                     float* __restrict__ S) {
    __shared__ float As[2][16 * AS_STRIDE];
    __shared__ float Bs[2][TILE_K * BS_STRIDE];

    const int tid  = threadIdx.x;
    const int wave = tid >> 5;        // 0..7
    const int lane = tid & 31;
    const int half = lane >> 4;       // 0 or 1 (lane group)
    const int l16  = lane & 15;
    const int m0   = blockIdx.x * 16; // 100000 / 16 = 6250 blocks exactly
    const int n0   = wave * 32;       // this wave's two N-tiles: n0, n0+16

    // staging assignments (fixed per thread)
    const int cA = tid & 31;          // K within chunk for A tile
    const int rA = tid >> 5;          // row 0..7 (and +8)

    v8f acc0 = {};
    v8f acc1 = {};

    float a_reg0, a_reg1;
    v4f   b_reg[8];

    // ---- prefetch chunk 0 into registers ----
    {
        const int k0 = 0;
        a_reg0 = __builtin_nontemporal_load(X + (size_t)(m0 + rA    ) * D + k0 + cA);
        a_reg1 = __builtin_nontemporal_load(X + (size_t)(m0 + rA + 8) * D + k0 + cA);
#pragma unroll
        for (int i = 0; i < 8; ++i) {
            const int idx4 = tid + 256 * i;
            const int r    = idx4 >> 6;
            const int c4   = idx4 & 63;
            b_reg[i] = *(const v4f*)(W + (size_t)(k0 + r) * D + c4 * 4);
        }
    }
    // ---- stage chunk 0 into buffer 0 ----
    {
        As[0][(rA    ) * AS_STRIDE + cA] = a_reg0;
        As[0][(rA + 8) * AS_STRIDE + cA] = a_reg1;
#pragma unroll
        for (int i = 0; i < 8; ++i) {
            const int idx4 = tid + 256 * i;
            const int r    = idx4 >> 6;
            const int c4   = idx4 & 63;
            *(v4f*)(&Bs[0][r * BS_STRIDE + c4 * 4]) = b_reg[i];
        }
    }
    __syncthreads();

    for (int chunk = 0; chunk < 8; ++chunk) {
        const int cur = chunk & 1;
        const int nxt = cur ^ 1;

        // ---- issue global loads for chunk+1 (latency hidden under WMMAs) ----
        if (chunk < 7) {
            const int k0 = (chunk + 1) * TILE_K;
            a_reg0 = __builtin_nontemporal_load(X + (size_t)(m0 + rA    ) * D + k0 + cA);
            a_reg1 = __builtin_nontemporal_load(X + (size_t)(m0 + rA + 8) * D + k0 + cA);
#pragma unroll
            for (int i = 0; i < 8; ++i) {
                const int idx4 = tid + 256 * i;
                const int r    = idx4 >> 6;
                const int c4   = idx4 & 63;
                b_reg[i] = *(const v4f*)(W + (size_t)(k0 + r) * D + c4 * 4);
            }
        }

        // ---- 8 k-substeps of 4, two 16x16 WMMAs each ----
#pragma unroll
        for (int s = 0; s < 8; ++s) {
            const int kb = s * 4 + half * 2;   // per-lane K base within chunk
            v2f a, b0, b1;
            a[0]  = As[cur][l16 * AS_STRIDE + kb + 0];
            a[1]  = As[cur][l16 * AS_STRIDE + kb + 1];
            b0[0] = Bs[cur][(kb + 0) * BS_STRIDE + n0 + l16];
            b0[1] = Bs[cur][(kb + 1) * BS_STRIDE + n0 + l16];
            b1[0] = Bs[cur][(kb + 0) * BS_STRIDE + n0 + 16 + l16];
            b1[1] = Bs[cur][(kb + 1) * BS_STRIDE + n0 + 16 + l16];
            acc0 = __builtin_amdgcn_wmma_f32_16x16x4_f32(false, a, false, b0,
                                                         (short)0, acc0, false, false);
            acc1 = __builtin_amdgcn_wmma_f32_16x16x4_f32(false, a, false, b1,
                                                         (short)0, acc1, false, false);
        }

        // ---- stage chunk+1 into the other buffer, then one barrier ----
        if (chunk < 7) {
            As[nxt][(rA    ) * AS_STRIDE + cA] = a_reg0;
            As[nxt][(rA + 8) * AS_STRIDE + cA] = a_reg1;
#pragma unroll
            for (int i = 0; i < 8; ++i) {
                const int idx4 = tid + 256 * i;
                const int r    = idx4 >> 6;
                const int c4   = idx4 & 63;
                *(v4f*)(&Bs[nxt][r * BS_STRIDE + c4 * 4]) = b_reg[i];
            }
        }
        __syncthreads();
    }

    // ---- write C: VGPR v -> row (half*8 + v), col = l16 within each N-tile ----
#pragma unroll
    for (int v = 0; v < 8; ++v) {
        const int row = m0 + half * 8 + v;
        S[(size_t)row * D + n0 +      l16] = acc0[v];
        S[(size_t)row * D + n0 + 16 + l16] = acc1[v];
    }
}

// ---------------------------------------------------------------------------
// SpMM: one wave per edge; lane owns 8 contiguous floats (2x b128 gather,
// 8x global_atomic_add_f32 scatter). support/out rows are L2-resident.
// ---------------------------------------------------------------------------
__global__ __launch_bounds__(256) void gcn_spmm(const float* __restrict__ S,
                                                const float* __restrict__ ev,
                                                const int* __restrict__ er,
                                                const int* __restrict__ ec,
                                                float* __restrict__ out, int nE) {
    const int wavesPerBlock = blockDim.x >> 5;
    const int e = blockIdx.x * wavesPerBlock + (threadIdx.x >> 5);
    if (e >= nE) return;
    const int lane = threadIdx.x & 31;

    const int   r = er[e];
    const int   c = ec[e];
    const float v = ev[e];

    const v4f* src = (const v4f*)(S + (size_t)c * D) + lane * 2;
    const v4f s0 = src[0];
    const v4f s1 = src[1];

    float* dst = out + (size_t)r * D + lane * 8;
    atomicAdd(dst + 0, v * s0[0]);
    atomicAdd(dst + 1, v * s0[1]);
    atomicAdd(dst + 2, v * s0[2]);
    atomicAdd(dst + 3, v * s0[3]);
    atomicAdd(dst + 4, v * s1[0]);
    atomicAdd(dst + 5, v * s1[1]);
    atomicAdd(dst + 6, v * s1[2]);
    atomicAdd(dst + 7, v * s1[3]);
}

// ---------------------------------------------------------------------------
__global__ __launch_bounds__(256) void gcn_zero(float* __restrict__ out, int n4) {
    const int i = blockIdx.x * blockDim.x + threadIdx.x;
    if (i < n4) {
        v4f z = {0.f, 0.f, 0.f, 0.f};
        ((v4f*)out)[i] = z;
    }
}

__global__ __launch_bounds__(256) void gcn_relu(float* __restrict__ out, int n4) {
    const int i = blockIdx.x * blockDim.x + threadIdx.x;
    if (i < n4) {
        v4f v = ((v4f*)out)[i];
        v[0] = fmaxf(v[0], 0.f); v[1] = fmaxf(v[1], 0.f);
        v[2] = fmaxf(v[2], 0.f); v[3] = fmaxf(v[3], 0.f);
        ((v4f*)out)[i] = v;
    }
}

// ---------------------------------------------------------------------------
extern "C" void kernel_launch(void* const* d_in, const int* in_sizes, int n_in,
                              void* d_out, int out_size, void* d_ws, size_t ws_size,
                              hipStream_t stream) {
    const float* x  = (const float*)d_in[0];   // [N, 256]
    const float* w  = (const float*)d_in[1];   // [256, 256]
    const float* ev = (const float*)d_in[2];   // [E]
    const int*   er = (const int*)  d_in[3];   // [E]
    const int*   ec = (const int*)  d_in[4];   // [E]
    float* out      = (float*)d_out;           // [N, 256]
    float* support  = (float*)d_ws;            // [N, 256] scratch

    const int nNodes = in_sizes[0] / D;        // 100000
    const int nE     = in_sizes[2];            // 3200000
    const int n4     = (nNodes * D) / 4;

    gcn_zero<<<(n4 + 255) / 256, 256, 0, stream>>>(out, n4);
    gcn_gemm_wmma<<<nNodes / 16, 256, 0, stream>>>(x, w, support);
    gcn_spmm<<<(nE + 7) / 8, 256, 0, stream>>>(support, ev, er, ec, out, nE);
    gcn_relu<<<(n4 + 255) / 256, 256, 0, stream>>>(out, n4);
}